// MMCL_30588757082558
// MI455X (gfx1250) — compile-verified
//
#include <hip/hip_runtime.h>
#include <math.h>

// ---------------- problem constants (from reference) ----------------
constexpr int   TPB   = 256;          // 8 wave32s per workgroup
constexpr int   NCOL  = 100000;       // N
constexpr int   NV4   = NCOL / 4;     // 25000 float4 per row
constexpr int   KPOS  = 4;
constexpr int   MNEG  = 999;          // int(0.01 * (N - K))
constexpr float SCALE = 10.0f;
constexpr float LAM   = 0.1f;

constexpr int HBITS = 12;             // global histogram: key[31:20]
constexpr int HBINS = 1 << HBITS;     // 4096 bins (16 KB)
constexpr int LBINS = 1024;           // refinement levels: 10 bits each
constexpr int CAPL  = 32;             // private candidate capacity per thread

// Monotone bit-key: ascending key order == ascending float order.
__device__ __forceinline__ unsigned fkey(float f) {
  unsigned u = __float_as_uint(f);
  unsigned m = (unsigned)((int)u >> 31) | 0x80000000u;
  return u ^ m;
}
// Inverse of fkey (exact reconstruction of the threshold value).
__device__ __forceinline__ float fkey_inv(unsigned key) {
  unsigned u = (key & 0x80000000u) ? (key ^ 0x80000000u) : ~key;
  return __uint_as_float(u);
}

// -------- CDNA5 async global->LDS copy (gfx1250) --------
__device__ __forceinline__ void async_ld_b128(unsigned lds_byte, unsigned long long gaddr) {
  asm volatile("global_load_async_to_lds_b128 %0, %1, off"
               :: "v"(lds_byte), "v"(gaddr) : "memory");
}
__device__ __forceinline__ void wait_async0() { asm volatile("s_wait_asynccnt 0" ::: "memory"); }
__device__ __forceinline__ void wait_async1() { asm volatile("s_wait_asynccnt 1" ::: "memory"); }

// Stream one row (NV4 float4s) through LDS with double-buffered async copies.
// Each thread consumes only the 16B slot it loaded itself -> only per-wave
// asynccnt waits needed. Issues are UNCONDITIONAL (tail addresses clamped to a
// valid element) so every wave has exactly one outstanding issue per step and
// `s_wait_asynccnt 1` is always the correct threshold.
template <class F>
__device__ __forceinline__ void stream_row(const float* __restrict__ rp,
                                           float (&tile)[2][TPB * 4],
                                           int tid, F f) {
  const unsigned long long base = (unsigned long long)rp;
  const unsigned lds0 = (unsigned)(size_t)(&tile[0][tid * 4]);
  const unsigned lds1 = (unsigned)(size_t)(&tile[1][tid * 4]);
  constexpr int NCH = (NV4 + TPB - 1) / TPB;   // 98 chunks
  auto gaddr = [&](int t) {
    int tc = t < NV4 ? t : NV4 - 1;            // clamp: redundant load, data unused
    return base + 16ull * (unsigned)tc;
  };
  async_ld_b128(lds0, gaddr(tid));
  for (int c = 0; c < NCH; ++c) {
    if (c + 1 < NCH) {
      async_ld_b128(((c + 1) & 1) ? lds1 : lds0, gaddr((c + 1) * TPB + tid));
      wait_async1();                           // chunk c complete (in-order per wave)
    } else {
      wait_async0();
    }
    int t = c * TPB + tid;
    if (t < NV4) {
      const float4 v = *(const float4*)&tile[c & 1][tid * 4];
      int g = 4 * t;
      f(v.x, g); f(v.y, g + 1); f(v.z, g + 2); f(v.w, g + 3);
    }
  }
}

// One workgroup per row: exact top-M exp-sum via histogram select, then loss.
// Fully deterministic: integer LDS atomics only for counts; every expf lands in
// a fixed thread's accumulator in a fixed order; tie value rebuilt from key bits.
__global__ __launch_bounds__(TPB) void mmcl_row_kernel(const float* __restrict__ logits,
                                                       const int*   __restrict__ targets,
                                                       float*       __restrict__ row_loss) {
  __shared__ unsigned hist[HBINS];                 // 16 KB (levels reuse first 1024)
  __shared__ alignas(16) float tile[2][TPB * 4];   //  8 KB async staging
  __shared__ float accbuf[TPB];                    //  1 KB
  __shared__ unsigned bsum[TPB];                   //  1 KB
  __shared__ int   s_bstar, s_need, s_sub, s_above;
  __shared__ float s_posdot;

  const int row = blockIdx.x;
  const int tid = threadIdx.x;
  const float* rp = logits + (size_t)row * NCOL;
  const int tgt = targets[row];
  const int p0 = tgt;
  const int p1 = (tgt +      NCOL / 4) % NCOL;
  const int p2 = (tgt + 2 * (NCOL / 4)) % NCOL;
  const int p3 = (tgt + 3 * (NCOL / 4)) % NCOL;

  for (int i = tid; i < HBINS; i += TPB) hist[i] = 0;
  __syncthreads();

  // ---- pass 1: 12-bit key histogram over all negatives ----
  stream_row(rp, tile, tid, [&](float x, int g) {
    if (g != p0 && g != p1 && g != p2 && g != p3)
      atomicAdd(&hist[fkey(x) >> (32 - HBITS)], 1u);
  });
  __syncthreads();

  // ---- find threshold bin b*: count(bins > b*) < M <= count(bins >= b*) ----
  { unsigned s = 0;
    for (int i = 0; i < HBINS / TPB; ++i) s += hist[tid * (HBINS / TPB) + i];
    bsum[tid] = s; }
  __syncthreads();
  if (tid == 0) {
    unsigned cum = 0;
    for (int blk = TPB - 1; blk >= 0; --blk) {
      if (cum + bsum[blk] >= (unsigned)MNEG) {
        for (int i = HBINS / TPB - 1; i >= 0; --i) {
          unsigned h = hist[blk * (HBINS / TPB) + i];
          if (cum + h >= (unsigned)MNEG) {
            s_bstar = blk * (HBINS / TPB) + i; s_need = MNEG - (int)cum; break;
          }
          cum += h;
        }
        break;
      }
      cum += bsum[blk];
    }
  }
  __syncthreads();
  const unsigned bstar = (unsigned)s_bstar;

  // ---- pass 2: exp-sum strictly above bin; private threshold-bin candidates ----
  float acc = 0.0f;
  float lbuf[CAPL];
  int   nloc = 0;
  stream_row(rp, tile, tid, [&](float x, int g) {
    if (g != p0 && g != p1 && g != p2 && g != p3) {
      unsigned b = fkey(x) >> (32 - HBITS);
      if (b > bstar) {
        acc += expf(SCALE * x);
      } else if (b == bstar) {
        if (nloc < CAPL) lbuf[nloc++] = x;     // per-thread, deterministic order
      }
    }
  });

  // ---- exact refinement of remaining 20 key bits: 2 levels x 10 bits ----
  int need = s_need;
  unsigned sub_hi = 0, sub_lo = 0;
  for (int L = 0; L < 2; ++L) {
    const int sh = (L == 0) ? 10 : 0;
    for (int i = tid; i < LBINS; i += TPB) hist[i] = 0;
    __syncthreads();
    for (int i = 0; i < nloc; ++i)
      atomicAdd(&hist[(fkey(lbuf[i]) >> sh) & (LBINS - 1)], 1u);
    __syncthreads();
    if (tid == 0) {
      unsigned cum = 0; int fs = 0; unsigned ab = 0;
      for (int s = LBINS - 1; s >= 0; --s) {
        unsigned h = hist[s];
        if (cum + h >= (unsigned)need) { fs = s; ab = cum; break; }
        cum += h;
      }
      s_sub = fs; s_above = (int)ab;
    }
    __syncthreads();
    const int subst = s_sub;
    need -= s_above;
    if (L == 0) {
      sub_hi = (unsigned)subst;
      int keep = 0;
      for (int i = 0; i < nloc; ++i) {
        float v = lbuf[i];
        int sub = (int)((fkey(v) >> 10) & (LBINS - 1));
        if (sub > subst) acc += expf(SCALE * v);
        else if (sub == subst) lbuf[keep++] = v;
      }
      nloc = keep;
    } else {
      sub_lo = (unsigned)subst;
      for (int i = 0; i < nloc; ++i) {
        float v = lbuf[i];
        int sub = (int)(fkey(v) & (LBINS - 1));
        if (sub > subst) acc += expf(SCALE * v);
      }
    }
  }

  // `need` ties remain, all exactly equal to the threshold value (key fully known).
  if (tid == 0) {
    unsigned key = (bstar << 20) | (sub_hi << 10) | sub_lo;
    float tval = fkey_inv(key);
    if (need > 0) acc += (float)need * expf(SCALE * tval);
    // positives + weighted positive-logit term
    const int pp[4] = { p0, p1, p2, p3 };
    float pd = 0.0f;
    for (int j = 0; j < KPOS; ++j) {
      float x = rp[pp[j]];
      acc += expf(SCALE * x);
      float w = (pp[j] == tgt) ? LAM : ((1.0f - LAM) / (KPOS - 1));
      pd += w * (SCALE * x);
    }
    s_posdot = pd;
  }
  accbuf[tid] = acc;
  __syncthreads();
  for (int s = TPB / 2; s > 0; s >>= 1) {
    if (tid < s) accbuf[tid] += accbuf[tid + s];
    __syncthreads();
  }
  if (tid == 0) row_loss[row] = logf(accbuf[0]) - s_posdot;
}

// ---- final mean over B rows: single wave32, sum via V_WMMA_F32_16X16X4_F32 ----
typedef __attribute__((ext_vector_type(2))) float v2f;
typedef __attribute__((ext_vector_type(8))) float v8f;

__global__ __launch_bounds__(32) void mmcl_reduce_kernel(const float* __restrict__ rl,
                                                         float* __restrict__ out, int B) {
  const int lane = threadIdx.x;
  float s = 0.0f;
  for (int i = lane; i < B; i += 32) s += rl[i];
  // A (16x4 f32): VGPR0 lanes0-15 -> (M=lane,K=0), lanes16-31 -> (M=lane-16,K=2); VGPR1 = 0.
  // B (4x16) all ones -> D[m][n] = s[m] + s[m+16] for every column n.
  v2f a; a.x = s;    a.y = 0.0f;
  v2f b; b.x = 1.0f; b.y = 1.0f;
  v8f c = {};
  c = __builtin_amdgcn_wmma_f32_16x16x4_f32(false, a, false, b, (short)0, c,
                                            false, false);
  float t = c[0] + c[1] + c[2] + c[3] + c[4] + c[5] + c[6] + c[7];
  t += __shfl_xor(t, 16, 32);      // lanes 0-15 hold rows 0-7; lanes 16-31 rows 8-15
  if (lane == 0) out[0] = t * (1.0f / (float)B);
}

extern "C" void kernel_launch(void* const* d_in, const int* in_sizes, int n_in,
                              void* d_out, int out_size, void* d_ws, size_t ws_size,
                              hipStream_t stream) {
  const float* logits  = (const float*)d_in[0];   // [B, N] f32
  const int*   targets = (const int*)d_in[1];     // [B] i32
  (void)n_in; (void)out_size; (void)ws_size;      // multi_targets (d_in[2]) derivable from targets
  const int B = in_sizes[1];
  float* row_loss = (float*)d_ws;                 // B floats of scratch

  mmcl_row_kernel<<<B, TPB, 0, stream>>>(logits, targets, row_loss);
  mmcl_reduce_kernel<<<1, 32, 0, stream>>>(row_loss, (float*)d_out, B);
}